// CrossGatNet_55104430408228
// MI455X (gfx1250) — compile-verified
//
#include <hip/hip_runtime.h>
#include <math.h>

typedef __attribute__((ext_vector_type(2))) float v2f;
typedef __attribute__((ext_vector_type(8))) float v8f;

#define FEAT 128
#define CPH 64
#define NEG_SLOPE 0.2f
#define ENC_NEG_INF 0x007FFFFFu   // f2ord(-inf)

static __device__ __forceinline__ unsigned f2ord(float f) {
    unsigned u = __float_as_uint(f);
    return (u & 0x80000000u) ? ~u : (u | 0x80000000u);
}
static __device__ __forceinline__ float ord2f(unsigned u) {
    unsigned b = (u & 0x80000000u) ? (u & 0x7fffffffu) : ~u;
    return __uint_as_float(b);
}

// ---------------- Node GEMM: Y[N,128] = X[N,128] @ W[128,128] ----------------
// One wave per 16x16 tile; 8 waves/block -> 16 rows x 128 cols per block.
__global__ void gemm128_wmma(const float* __restrict__ X, const float* __restrict__ W,
                             float* __restrict__ Y, int N) {
    const int wave = threadIdx.x >> 5;
    const int lane = threadIdx.x & 31;
    const int half = lane >> 4;      // 0: K=0..1, 1: K=2..3 (A & B frag layout)
    const int lr   = lane & 15;
    const int row0 = blockIdx.x * 16;
    const int col0 = wave * 16;
    int arow = row0 + lr; if (arow >= N) arow = N - 1;   // clamp loads, keep EXEC full
    const int bcol = col0 + lr;
    v8f acc = {};
#pragma unroll
    for (int k = 0; k < FEAT; k += 4) {
        const int ka = k + 2 * half;
        v2f a, b;
        a.x = X[arow * FEAT + ka];
        a.y = X[arow * FEAT + ka + 1];
        b.x = W[ka * FEAT + bcol];
        b.y = W[(ka + 1) * FEAT + bcol];
        acc = __builtin_amdgcn_wmma_f32_16x16x4_f32(false, a, false, b, (short)0, acc,
                                                    false, false);
    }
#pragma unroll
    for (int i = 0; i < 8; ++i) {
        int r = row0 + i + 8 * half;
        if (r < N) Y[r * FEAT + col0 + lr] = acc[i];
    }
}

// ---------------- Per-node attention logits a[n,h] = <x[n,h,:], att[h,:]> ----------------
__global__ void attn_logits(const float* __restrict__ X, const float* __restrict__ att,
                            float* __restrict__ A, int N) {
    int i = blockIdx.x * blockDim.x + threadIdx.x;
    if (i >= N * 2) return;
    int n = i >> 1, h = i & 1;
    const float* xp = X + (size_t)n * FEAT + h * CPH;
    const float* ap = att + h * CPH;
    float s = 0.f;
#pragma unroll 8
    for (int c = 0; c < CPH; ++c) s += xp[c] * ap[c];
    A[i] = s;
}

// ---------------- Init accumulators ----------------
__global__ void init_bufs(float* __restrict__ xconv, unsigned* __restrict__ emax,
                          float* __restrict__ denom, int Nt) {
    size_t i = (size_t)blockIdx.x * blockDim.x + threadIdx.x;
    if (i < (size_t)Nt * FEAT) xconv[i] = 0.f;
    if (i < (size_t)Nt * 2) { emax[i] = ENC_NEG_INF; denom[i] = 0.f; }
}

// ---------------- Segment max over edges (monotone-u32 atomicMax) ----------------
__global__ void edge_max(const int* __restrict__ ei, const float* __restrict__ a_s,
                         const float* __restrict__ a_t, unsigned* __restrict__ emax, int E) {
    int i = blockIdx.x * blockDim.x + threadIdx.x;
    if (i >= E * 2) return;
    int e = i >> 1, h = i & 1;
    int s = ei[e], t = ei[E + e];
    float v = a_s[s * 2 + h] + a_t[t * 2 + h];
    v = v > 0.f ? v : NEG_SLOPE * v;
    atomicMax(&emax[t * 2 + h], f2ord(v));
}

// ---------------- exp(e - max) + segment sum ----------------
__global__ void edge_exp(const int* __restrict__ ei, const float* __restrict__ a_s,
                         const float* __restrict__ a_t, const unsigned* __restrict__ emax,
                         float* __restrict__ wbuf, float* __restrict__ denom, int E) {
    int i = blockIdx.x * blockDim.x + threadIdx.x;
    if (i >= E * 2) return;
    int e = i >> 1, h = i & 1;
    int s = ei[e], t = ei[E + e];
    float v = a_s[s * 2 + h] + a_t[t * 2 + h];
    v = v > 0.f ? v : NEG_SLOPE * v;
    float m = ord2f(emax[t * 2 + h]);
    if (!isfinite(m)) m = 0.f;
    float w = __expf(v - m);
    wbuf[i] = w;
    atomicAdd(&denom[t * 2 + h], w);
}

// ---------------- Message scatter: xconv[tgt] += xs[src] * alpha ----------------
// One wave per edge; each lane handles 4 consecutive channels (same head).
__global__ void scatter_msg(const int* __restrict__ ei, const float* __restrict__ xs,
                            const float* __restrict__ wbuf, const float* __restrict__ denom,
                            float* __restrict__ xconv, int E) {
    size_t gid = (size_t)blockIdx.x * blockDim.x + threadIdx.x;
    int e = (int)(gid >> 5);
    int lane = (int)(gid & 31);
    if (e >= E) return;
    int s = ei[e], t = ei[E + e];
    float al0 = wbuf[e * 2 + 0] / (denom[t * 2 + 0] + 1e-16f);
    float al1 = wbuf[e * 2 + 1] / (denom[t * 2 + 1] + 1e-16f);
    int c0 = lane * 4;
    float al = (c0 < CPH) ? al0 : al1;
    const float4 xv = *(const float4*)(xs + (size_t)s * FEAT + c0);
    float* dst = xconv + (size_t)t * FEAT + c0;
    atomicAdd(dst + 0, xv.x * al);
    atomicAdd(dst + 1, xv.y * al);
    atomicAdd(dst + 2, xv.z * al);
    atomicAdd(dst + 3, xv.w * al);
}

// ---------------- Fused epilogue: gate/tanh GEMMs + mix ----------------
// accg = (x+b) @ WgX + tgt @ WgT ; accf = (x+b) @ Wf
// out = sigmoid(accg)*tanh(accf) + (1-sigmoid(accg))*tgt
__global__ void fuse_out(const float* __restrict__ xconv, const float* __restrict__ bias,
                         const float* __restrict__ tgt, const float* __restrict__ Wg,
                         const float* __restrict__ Wf, const int* __restrict__ dirp,
                         float* __restrict__ out, int Nt) {
    const int dir  = dirp[0];
    const int wave = threadIdx.x >> 5;
    const int lane = threadIdx.x & 31;
    const int half = lane >> 4;
    const int lr   = lane & 15;
    const int row0 = blockIdx.x * 16;
    const int col0 = wave * 16;
    int arow = row0 + lr; if (arow >= Nt) arow = Nt - 1;
    const int bcol = col0 + lr;
    // direction 0: h = [x, tgt]  -> x uses Wg rows 0..127, tgt uses rows 128..255
    const float* WgX = Wg + (dir == 0 ? 0 : FEAT * FEAT);
    const float* WgT = Wg + (dir == 0 ? FEAT * FEAT : 0);
    v8f accg = {}, accf = {};
#pragma unroll
    for (int k = 0; k < FEAT; k += 4) {
        const int ka = k + 2 * half;
        v2f ax, at, bg1, bg2, bf;
        ax.x = xconv[(size_t)arow * FEAT + ka]     + bias[ka];
        ax.y = xconv[(size_t)arow * FEAT + ka + 1] + bias[ka + 1];
        at.x = tgt[(size_t)arow * FEAT + ka];
        at.y = tgt[(size_t)arow * FEAT + ka + 1];
        bg1.x = WgX[ka * FEAT + bcol]; bg1.y = WgX[(ka + 1) * FEAT + bcol];
        bg2.x = WgT[ka * FEAT + bcol]; bg2.y = WgT[(ka + 1) * FEAT + bcol];
        bf.x  = Wf[ka * FEAT + bcol];  bf.y  = Wf[(ka + 1) * FEAT + bcol];
        accg = __builtin_amdgcn_wmma_f32_16x16x4_f32(false, ax, false, bg1, (short)0, accg,
                                                     false, false);
        accg = __builtin_amdgcn_wmma_f32_16x16x4_f32(false, at, false, bg2, (short)0, accg,
                                                     false, false);
        accf = __builtin_amdgcn_wmma_f32_16x16x4_f32(false, ax, false, bf, (short)0, accf,
                                                     false, false);
    }
#pragma unroll
    for (int i = 0; i < 8; ++i) {
        int r = row0 + i + 8 * half;
        if (r < Nt) {
            float tv = tgt[(size_t)r * FEAT + col0 + lr];
            float g  = 1.f / (1.f + __expf(-accg[i]));
            out[(size_t)r * FEAT + col0 + lr] = g * tanhf(accf[i]) + (1.f - g) * tv;
        }
    }
}

extern "C" void kernel_launch(void* const* d_in, const int* in_sizes, int n_in,
                              void* d_out, int out_size, void* d_ws, size_t ws_size,
                              hipStream_t stream) {
    const float* src    = (const float*)d_in[0];
    const float* tgt    = (const float*)d_in[1];
    const float* W_src  = (const float*)d_in[2];
    const float* W_tgt  = (const float*)d_in[3];
    const float* att_s  = (const float*)d_in[4];
    const float* att_d  = (const float*)d_in[5];
    const float* bias   = (const float*)d_in[6];
    const float* Wg     = (const float*)d_in[7];
    const float* Wf     = (const float*)d_in[8];
    const int*   ei     = (const int*)d_in[9];
    const int*   dirp   = (const int*)d_in[10];

    const int Ns = in_sizes[0] / FEAT;
    const int Nt = in_sizes[1] / FEAT;
    const int E  = in_sizes[9] / 2;

    // workspace layout (floats)
    float* ws    = (float*)d_ws;
    float* xs    = ws;                                   // Ns*128
    float* xt    = xs    + (size_t)Ns * FEAT;            // Nt*128
    float* xconv = xt    + (size_t)Nt * FEAT;            // Nt*128
    float* a_s   = xconv + (size_t)Nt * FEAT;            // Ns*2
    float* a_t   = a_s   + (size_t)Ns * 2;               // Nt*2
    unsigned* emax = (unsigned*)(a_t + (size_t)Nt * 2);  // Nt*2
    float* denom = (float*)(emax + (size_t)Nt * 2);      // Nt*2
    float* wbuf  = denom + (size_t)Nt * 2;               // E*2

    dim3 blk(256);
    gemm128_wmma<<<(Ns + 15) / 16, blk, 0, stream>>>(src, W_src, xs, Ns);
    gemm128_wmma<<<(Nt + 15) / 16, blk, 0, stream>>>(tgt, W_tgt, xt, Nt);
    attn_logits<<<(Ns * 2 + 255) / 256, blk, 0, stream>>>(xs, att_s, a_s, Ns);
    attn_logits<<<(Nt * 2 + 255) / 256, blk, 0, stream>>>(xt, att_d, a_t, Nt);
    init_bufs<<<(int)(((size_t)Nt * FEAT + 255) / 256), blk, 0, stream>>>(xconv, emax, denom, Nt);
    edge_max<<<(E * 2 + 255) / 256, blk, 0, stream>>>(ei, a_s, a_t, emax, E);
    edge_exp<<<(E * 2 + 255) / 256, blk, 0, stream>>>(ei, a_s, a_t, emax, wbuf, denom, E);
    scatter_msg<<<(int)(((size_t)E * 32 + 255) / 256), blk, 0, stream>>>(ei, xs, wbuf, denom, xconv, E);
    fuse_out<<<(Nt + 15) / 16, blk, 0, stream>>>(xconv, bias, tgt, Wg, Wf, dirp, (float*)d_out, Nt);
}